// SA_27668179321328
// MI455X (gfx1250) — compile-verified
//
#include <hip/hip_runtime.h>
#include <hip/hip_bf16.h>
#include <math.h>

// ---------------------------------------------------------------------------
// ST non-local attention block on gfx1250 (CDNA5, wave32).
// Heavy GEMMs (1x1x1 qkv projection + two 3x3x3 convs as implicit GEMM) run
// on V_WMMA_F32_16X16X4_F32 (exact f32 matrix pipe). Softmax branches are
// VALU (v_exp_f32). BN folded to per-channel alpha/beta in epilogues.
// ---------------------------------------------------------------------------

typedef __attribute__((ext_vector_type(2))) float v2f;
typedef __attribute__((ext_vector_type(8))) float v8f;

#define BN_EPS 1e-5f
#define NB    2
#define C_IN  512
#define CC    64
#define DHW   2048   // 8*16*16
#define KTAPS 27
#define KPACK 1728   // CC*27

__device__ __forceinline__ v8f wmma4(v2f a, v2f b, v8f c) {
  // D = A(16x4 f32) * B(4x16 f32) + C(16x16 f32)
  return __builtin_amdgcn_wmma_f32_16x16x4_f32(
      /*neg_a=*/false, a, /*neg_b=*/false, b,
      /*c_mod=*/(short)0, c, /*reuse_a=*/false, /*reuse_b=*/false);
}

// ---- fold BN (+conv bias) into y = relu(conv*alpha + beta) ------------------
__global__ void bn_prep_kernel(const float* __restrict__ b, const float* __restrict__ g,
                               const float* __restrict__ bt, const float* __restrict__ m,
                               const float* __restrict__ v, float* __restrict__ alpha,
                               float* __restrict__ beta, int n) {
  int i = blockIdx.x * blockDim.x + threadIdx.x;
  if (i < n) {
    float a = g[i] * rsqrtf(v[i] + BN_EPS);
    alpha[i] = a;
    beta[i]  = (b[i] - m[i]) * a + bt[i];
  }
}

// ---- repack conv weights [co][ci][t] -> [co][t][ci] (tap-major K order) -----
__global__ void pack_w_kernel(const float* __restrict__ w, float* __restrict__ wp) {
  int idx = blockIdx.x * blockDim.x + threadIdx.x;   // C_IN*KPACK elements
  if (idx < C_IN * KPACK) {
    int co = idx / KPACK;
    int r  = idx - co * KPACK;
    int t  = r >> 6;           // r / 64
    int ci = r & 63;
    wp[idx] = w[co * KPACK + ci * KTAPS + t];
  }
}

// ---- q/k/v 1x1x1 conv as GEMM: one wave -> 64ch x 16pos for all 3 proj ------
__global__ void __launch_bounds__(256) qkv_kernel(
    const float* __restrict__ x,
    const float* __restrict__ wq, const float* __restrict__ wk, const float* __restrict__ wv,
    const float* __restrict__ bnab,            // [qA64 qB64 kA64 kB64 vA64 vB64]
    float* __restrict__ qo, float* __restrict__ ko, float* __restrict__ vo) {
  int wid  = threadIdx.x >> 5;
  int lane = threadIdx.x & 31;
  int gw   = blockIdx.x * 8 + wid;             // 0..255 position tiles
  int n    = gw >> 7;
  int pb   = (gw & 127) << 4;                  // 16 consecutive positions
  int j    = lane & 15;
  int kh   = lane >> 4;                        // K half (0/1)

  const float* xb = x + (size_t)n * C_IN * DHW + pb + j;
  const float* wpt[3] = {wq, wk, wv};

  v8f z = {0.f,0.f,0.f,0.f,0.f,0.f,0.f,0.f};
  v8f acc[3][4];
#pragma unroll
  for (int p = 0; p < 3; ++p)
#pragma unroll
    for (int mt = 0; mt < 4; ++mt) acc[p][mt] = z;

  for (int k0 = 0; k0 < C_IN; k0 += 4) {
    int kb = k0 + 2 * kh;
    v2f b;
    b.x = xb[(size_t)kb * DHW];
    b.y = xb[(size_t)(kb + 1) * DHW];
#pragma unroll
    for (int p = 0; p < 3; ++p) {
      const float* wp = wpt[p];
#pragma unroll
      for (int mt = 0; mt < 4; ++mt) {
        v2f a = *(const v2f*)(wp + (size_t)(mt * 16 + j) * C_IN + kb);
        acc[p][mt] = wmma4(a, b, acc[p][mt]);
      }
    }
  }

  float* outs[3] = {qo, ko, vo};
#pragma unroll
  for (int p = 0; p < 3; ++p) {
    const float* al = bnab + p * 128;
    const float* be = al + 64;
    float* dst = outs[p] + (size_t)n * CC * DHW + pb + j;
#pragma unroll
    for (int mt = 0; mt < 4; ++mt)
#pragma unroll
      for (int r = 0; r < 8; ++r) {
        int co = mt * 16 + r + 8 * kh;
        float y = fmaxf(acc[p][mt][r] * al[co] + be[co], 0.f);
        dst[(size_t)co * DHW] = y;
      }
  }
}

// ---- 3x3x3 conv (Cin=64 -> Cout=512, pad 1) as implicit GEMM on WMMA --------
// one wave: 64 out-channels x 16 positions (one full W-row), K = 27 taps x 64ci
__global__ void __launch_bounds__(256) conv3_kernel(
    const float* __restrict__ in,    // [N][64][D][H][W]
    const float* __restrict__ wp,    // packed [512][27][64]
    const float* __restrict__ alpha, const float* __restrict__ beta,  // [512]
    float* __restrict__ dst) {       // [N][512][D][H][W]
  int wid  = threadIdx.x >> 5;
  int lane = threadIdx.x & 31;
  int gw   = blockIdx.x * 8 + wid;   // 0..2047
  int mg   = gw & 7;                 // out-channel group (64 wide)
  int tp   = gw >> 3;                // 0..255 position tiles
  int n    = tp >> 7;
  int d    = (tp >> 4) & 7;
  int h    = tp & 15;
  int j    = lane & 15;
  int kh   = lane >> 4;
  int cobase = mg * 64;

  const float* inb = in + (size_t)n * CC * DHW;

  v8f z = {0.f,0.f,0.f,0.f,0.f,0.f,0.f,0.f};
  v8f acc[4] = {z, z, z, z};

  for (int t = 0; t < KTAPS; ++t) {
    int dd = t / 9;
    int rr = t - dd * 9;
    int dh = rr / 3;
    int dw = rr - dh * 3;
    int dz = d + dd - 1;
    int dy = h + dh - 1;
    int dx = j + dw - 1;
    bool ok = ((unsigned)dz < 8u) && ((unsigned)dy < 16u) && ((unsigned)dx < 16u);
    int soff = ok ? (dz * 256 + dy * 16 + dx) : 0;     // safe address for masked lanes
    const float* bp = inb + soff;
    const float* ap = wp + t * 64;

    for (int ci0 = 0; ci0 < 64; ci0 += 4) {
      int kb = ci0 + 2 * kh;
      float bx = bp[(size_t)kb * DHW];
      float by = bp[(size_t)(kb + 1) * DHW];
      v2f b;
      b.x = ok ? bx : 0.f;
      b.y = ok ? by : 0.f;
#pragma unroll
      for (int mt = 0; mt < 4; ++mt) {
        v2f a = *(const v2f*)(ap + (size_t)(cobase + mt * 16 + j) * KPACK + kb);
        acc[mt] = wmma4(a, b, acc[mt]);
      }
    }
  }

  float* db = dst + (size_t)n * C_IN * DHW + d * 256 + h * 16 + j;
#pragma unroll
  for (int mt = 0; mt < 4; ++mt)
#pragma unroll
    for (int r = 0; r < 8; ++r) {
      int co = cobase + mt * 16 + r + 8 * kh;
      float y = fmaxf(acc[mt][r] * alpha[co] + beta[co], 0.f);
      db[(size_t)co * DHW] = y;
    }
}

// ---- spatial attention: per (n,c,d), rank-1 softmax over HW=256 -------------
__global__ void __launch_bounds__(256) patt_kernel(const float* __restrict__ q,
                                                   const float* __restrict__ k,
                                                   const float* __restrict__ v,
                                                   float* __restrict__ out) {
  __shared__ float sq[256], sk[256], sv[256], sw[256], sm[256];
  int tid = threadIdx.x;
  size_t base = (size_t)blockIdx.x * 256;
  sq[tid] = q[base + tid];
  sk[tid] = k[base + tid];
  sv[tid] = v[base + tid];
  __syncthreads();

  // pass 1: thread i -> row max, Z_i, weight w_i = v_i / Z_i
  float qi = sq[tid];
  float mx = -3.402823e38f;
  for (int jj = 0; jj < 256; ++jj) mx = fmaxf(mx, qi * sk[jj]);
  float Z = 0.f;
  for (int jj = 0; jj < 256; ++jj) Z += __expf(qi * sk[jj] - mx);
  sw[tid] = sv[tid] / Z;
  sm[tid] = mx;
  __syncthreads();

  // pass 2: thread j -> out[j] = sum_i w_i * exp(q_i*k_j - m_i)
  float kj = sk[tid];
  float o = 0.f;
  for (int ii = 0; ii < 256; ++ii) o += sw[ii] * __expf(sq[ii] * kj - sm[ii]);
  out[base + tid] = o;
}

// ---- temporal attention: per (n,c,h,w) over D=8 -----------------------------
__global__ void __launch_bounds__(256) datt_kernel(const float* __restrict__ q,
                                                   const float* __restrict__ k,
                                                   const float* __restrict__ v,
                                                   float* __restrict__ out) {
  int idx = blockIdx.x * blockDim.x + threadIdx.x;   // N*CC*HW = 32768
  if (idx >= NB * CC * 256) return;
  int nc = idx >> 8;
  int hw = idx & 255;
  size_t base = (size_t)nc * DHW + hw;

  float qd[8], kd[8], vd[8], o[8];
#pragma unroll
  for (int dl = 0; dl < 8; ++dl) {
    qd[dl] = q[base + dl * 256];
    kd[dl] = k[base + dl * 256];
    vd[dl] = v[base + dl * 256];
    o[dl] = 0.f;
  }
#pragma unroll
  for (int i = 0; i < 8; ++i) {
    float mx = -3.402823e38f;
#pragma unroll
    for (int jj = 0; jj < 8; ++jj) mx = fmaxf(mx, qd[i] * kd[jj]);
    float e[8];
    float Z = 0.f;
#pragma unroll
    for (int jj = 0; jj < 8; ++jj) { e[jj] = __expf(qd[i] * kd[jj] - mx); Z += e[jj]; }
    float wi = vd[i] / Z;
#pragma unroll
    for (int jj = 0; jj < 8; ++jj) o[jj] += wi * e[jj];
  }
#pragma unroll
  for (int jj = 0; jj < 8; ++jj) out[base + jj * 256] = o[jj];
}

// ---- final: out = gama*(Pc + Dc) + x ----------------------------------------
__global__ void combine_kernel(const float* __restrict__ pc, const float* __restrict__ dc,
                               const float* __restrict__ x, const float* __restrict__ gama,
                               float* __restrict__ out, int n) {
  int i = blockIdx.x * blockDim.x + threadIdx.x;
  if (i < n) out[i] = gama[0] * (pc[i] + dc[i]) + x[i];
}

// ---------------------------------------------------------------------------
extern "C" void kernel_launch(void* const* d_in, const int* in_sizes, int n_in,
                              void* d_out, int out_size, void* d_ws, size_t ws_size,
                              hipStream_t stream) {
  const float* x    = (const float*)d_in[0];
  const float* gama = (const float*)d_in[1];
  const float* Wq = (const float*)d_in[2];
  const float* bq = (const float*)d_in[3];
  const float* gq = (const float*)d_in[4];
  const float* btq= (const float*)d_in[5];
  const float* mq = (const float*)d_in[6];
  const float* vq = (const float*)d_in[7];
  const float* Wk = (const float*)d_in[8];
  const float* bk = (const float*)d_in[9];
  const float* gk = (const float*)d_in[10];
  const float* btk= (const float*)d_in[11];
  const float* mk = (const float*)d_in[12];
  const float* vk = (const float*)d_in[13];
  const float* Wv = (const float*)d_in[14];
  const float* bv = (const float*)d_in[15];
  const float* gv = (const float*)d_in[16];
  const float* btv= (const float*)d_in[17];
  const float* mv = (const float*)d_in[18];
  const float* vv = (const float*)d_in[19];
  const float* Ws = (const float*)d_in[20];
  const float* bs = (const float*)d_in[21];
  const float* gs = (const float*)d_in[22];
  const float* bts= (const float*)d_in[23];
  const float* ms = (const float*)d_in[24];
  const float* vs = (const float*)d_in[25];
  const float* Wd = (const float*)d_in[26];
  const float* bd = (const float*)d_in[27];
  const float* gd = (const float*)d_in[28];
  const float* btd= (const float*)d_in[29];
  const float* md = (const float*)d_in[30];
  const float* vd = (const float*)d_in[31];

  float* ws = (float*)d_ws;
  const size_t QKV = (size_t)NB * CC * DHW;        // 262144
  float* qx   = ws;
  float* kx   = qx + QKV;
  float* vx   = kx + QKV;
  float* patt = vx + QKV;
  float* datt = patt + QKV;
  float* pc   = datt + QKV;                        // 2097152 each
  float* dc   = pc + (size_t)NB * C_IN * DHW;
  float* wps  = dc + (size_t)NB * C_IN * DHW;      // 884736 each
  float* wpd  = wps + (size_t)C_IN * KPACK;
  float* bnqkv= wpd + (size_t)C_IN * KPACK;        // 384
  float* bnsA = bnqkv + 384;
  float* bnsB = bnsA + 512;
  float* bndA = bnsB + 512;
  float* bndB = bndA + 512;

  // BN folding
  bn_prep_kernel<<<1, 256, 0, stream>>>(bq, gq, btq, mq, vq, bnqkv,       bnqkv + 64,  64);
  bn_prep_kernel<<<1, 256, 0, stream>>>(bk, gk, btk, mk, vk, bnqkv + 128, bnqkv + 192, 64);
  bn_prep_kernel<<<1, 256, 0, stream>>>(bv, gv, btv, mv, vv, bnqkv + 256, bnqkv + 320, 64);
  bn_prep_kernel<<<2, 256, 0, stream>>>(bs, gs, bts, ms, vs, bnsA, bnsB, 512);
  bn_prep_kernel<<<2, 256, 0, stream>>>(bd, gd, btd, md, vd, bndA, bndB, 512);

  // weight repack for tap-major implicit GEMM
  pack_w_kernel<<<3456, 256, 0, stream>>>(Ws, wps);
  pack_w_kernel<<<3456, 256, 0, stream>>>(Wd, wpd);

  // q/k/v projections (WMMA)
  qkv_kernel<<<32, 256, 0, stream>>>(x, Wq, Wk, Wv, bnqkv, qx, kx, vx);

  // attention branches
  patt_kernel<<<NB * CC * 8, 256, 0, stream>>>(qx, kx, vx, patt);   // 1024 blocks
  datt_kernel<<<128, 256, 0, stream>>>(qx, kx, vx, datt);

  // 3x3x3 convs (WMMA implicit GEMM) + fused BN/ReLU
  conv3_kernel<<<256, 256, 0, stream>>>(patt, wps, bnsA, bnsB, pc);
  conv3_kernel<<<256, 256, 0, stream>>>(datt, wpd, bndA, bndB, dc);

  // residual combine
  combine_kernel<<<8192, 256, 0, stream>>>(pc, dc, x, gama, (float*)d_out,
                                           NB * C_IN * DHW);
}